// kernel_generated_1_21887153341238
// MI455X (gfx1250) — compile-verified
//
#include <hip/hip_runtime.h>
#include <hip/hip_bf16.h>

typedef float v2f __attribute__((ext_vector_type(2)));
typedef float v8f __attribute__((ext_vector_type(8)));

// ---------------------------------------------------------------------------
// Sizes (fixed by the reference problem)
// ---------------------------------------------------------------------------
#define H 28
#define W 28
#define CG 32           // grouped channels (64 -> 32 x 2)
#define MDIM 128        // output channels i
#define KDIM 576        // c(32)*kh(3)*kw(3)*l(2)
#define K2   (KDIM/2)   // 288 K-pairs
#define NDIM 784        // o(28)*p(28)

// Workspace layout (floats):
//   T5  : [c][kh][o][w][l]                    -> 150528
//   Zp  : [k2][n][2]   (K pair-interleaved)   -> 576*784 = 451584
//   W0p : [k2][i][2]   (K pair-interleaved)   -> 576*128 = 73728
#define T5_OFF  0
#define ZP_OFF  150528
#define W0P_OFF (150528 + 451584)

// ---------------------------------------------------------------------------
// Stage 1: t5[c,kh,o,w,l] = sum_g W1[2c+g,kh,l] * xroll[2c+g, o+kh-1, w]
//   xroll[j,h,w] = (0<=h<28) ? x[j,(h-1) mod 28, w] : 0
// ---------------------------------------------------------------------------
__global__ __launch_bounds__(256) void build_t5(const float* __restrict__ x,
                                                const float* __restrict__ W1,
                                                float* __restrict__ T5) {
    int idx = blockIdx.x * 256 + threadIdx.x;
    if (idx >= CG * 3 * H * W * 2) return;
    int l  = idx & 1;
    int r  = idx >> 1;
    int w  = r % W;  r /= W;
    int o  = r % H;  r /= H;
    int kh = r % 3;
    int c  = r / 3;

    float val = 0.0f;
    int h = o + kh - 1;
    if (h >= 0 && h < H) {
        int hs = (h + H - 1) % H;           // roll by +1 along H
        #pragma unroll
        for (int g = 0; g < 2; ++g) {
            int j = 2 * c + g;
            val += x[(j * H + hs) * W + w] * W1[(j * 3 + kh) * 2 + l];
        }
    }
    T5[idx] = val;
}

// ---------------------------------------------------------------------------
// Stage 2a: Zp[k2][n][2] = Z[2*k2 + k1][n]   (im2col gather with W-roll)
//   k = ((c*3+kh)*3+kw)*2 + l ; n = o*28 + p
//   q = p+kw-1 ; in range -> Z = T5[c,kh,o,(q-1) mod 28, l] ; else 0
// Flat output index: (k2*784 + n)*2 + k1  -> decode so writes are coalesced.
// ---------------------------------------------------------------------------
__global__ __launch_bounds__(256) void build_zp(const float* __restrict__ T5,
                                                float* __restrict__ Zp) {
    int idx = blockIdx.x * 256 + threadIdx.x;
    if (idx >= KDIM * NDIM) return;
    int k1 = idx & 1;
    int r  = idx >> 1;
    int n  = r % NDIM;
    int k2 = r / NDIM;
    int k  = k2 * 2 + k1;

    int p = n % W;
    int o = n / W;
    int l  = k & 1;
    int kk = k >> 1;
    int kw = kk % 3; kk /= 3;
    int kh = kk % 3;
    int c  = kk / 3;

    float val = 0.0f;
    int q = p + kw - 1;
    if (q >= 0 && q < W) {
        int wsrc = (q + W - 1) % W;         // roll by +1 along W
        val = T5[((((c * 3 + kh) * H + o) * W + wsrc) << 1) + l];
    }
    Zp[idx] = val;
}

// ---------------------------------------------------------------------------
// Stage 2b: W0p[k2][i][2] = W0[i][2*k2 + k1]  (W0 viewed as [128][576])
// Flat output index: (k2*128 + i)*2 + k1 -> coalesced writes.
// ---------------------------------------------------------------------------
__global__ __launch_bounds__(256) void build_w0p(const float* __restrict__ W0,
                                                 float* __restrict__ W0p) {
    int idx = blockIdx.x * 256 + threadIdx.x;
    if (idx >= KDIM * MDIM) return;
    int k1 = idx & 1;
    int r  = idx >> 1;
    int i  = r % MDIM;
    int k2 = r / MDIM;
    W0p[idx] = W0[i * KDIM + (k2 * 2 + k1)];
}

// ---------------------------------------------------------------------------
// Stage 2c: y[i][n] = sum_k W0[i][k] * Z[k][n]  via V_WMMA_F32_16X16X4_F32.
// Grid: 49 blocks (16-wide N tiles), 256 threads = 8 waves (16-row M tiles).
// A 16x4 f32 lane layout: lane%16 = M row; lane/16 selects K-pair {0,1}/{2,3}.
// B 4x16  f32 lane layout: lane%16 = N col; lane/16 selects K rows {0,1}/{2,3}.
// D 16x16 f32: VGPR r holds M = r + 8*(lane/16), N = lane%16.
// Pair-interleaved layouts make each fragment a single b64 load per lane.
// ---------------------------------------------------------------------------
__global__ __launch_bounds__(256) void gemm_wmma(const float* __restrict__ W0p,
                                                 const float* __restrict__ Zp,
                                                 float* __restrict__ y) {
    __shared__ __align__(16) float sZ[K2 * 16 * 2]; // 36,864 B: Z tile [k2][col][2]

    const int tid = threadIdx.x;
    const int n0  = blockIdx.x * 16;

    // Cooperative, coalesced b64 fill of Zp[:, n0:n0+16, :]
    for (int idx = tid; idx < K2 * 16; idx += 256) {
        int k2  = idx >> 4;
        int col = idx & 15;
        v2f v = *(const v2f*)(Zp + ((size_t)(k2 * NDIM + n0 + col) << 1));
        *(v2f*)(sZ + ((k2 * 16 + col) << 1)) = v;
    }
    __syncthreads();

    const int wave = tid >> 5;              // 0..7 -> M tile
    const int lane = tid & 31;
    const int col  = lane & 15;             // N col (B/D) and M row (A)
    const int half = lane >> 4;             // K/M half selector
    const int mRow = wave * 16 + col;       // global M row for A fetch

    const float* Abase = W0p + ((size_t)(half * MDIM + mRow) << 1);
    const float* Bbase = sZ + ((half * 16 + col) << 1);

    v8f acc = {};
    #pragma unroll 4
    for (int k2 = 0; k2 < K2; k2 += 2) {    // two K per pair -> K step 4
        v2f a = *(const v2f*)(Abase + ((size_t)k2 * MDIM << 1));
        v2f b = *(const v2f*)(Bbase + (k2 << 5));
        acc = __builtin_amdgcn_wmma_f32_16x16x4_f32(
            /*neg_a=*/false, a, /*neg_b=*/false, b,
            /*c_mod=*/(short)0, acc, /*reuse_a=*/false, /*reuse_b=*/false);
    }

    #pragma unroll
    for (int r = 0; r < 8; ++r) {
        int i = wave * 16 + r + 8 * half;   // D-matrix row mapping
        y[i * NDIM + n0 + col] = acc[r];
    }
}

// ---------------------------------------------------------------------------
extern "C" void kernel_launch(void* const* d_in, const int* in_sizes, int n_in,
                              void* d_out, int out_size, void* d_ws, size_t ws_size,
                              hipStream_t stream) {
    const float* x  = (const float*)d_in[0];   // (1,64,28,28)   = 50176
    const float* W0 = (const float*)d_in[1];   // (128,32,3,3,2) = 73728
    const float* W1 = (const float*)d_in[2];   // (64,3,2)       = 384
    float* y  = (float*)d_out;                 // (1,128,28,28)  = 100352
    float* ws = (float*)d_ws;

    float* T5  = ws + T5_OFF;
    float* Zp  = ws + ZP_OFF;
    float* W0p = ws + W0P_OFF;

    const int nT5 = CG * 3 * H * W * 2;        // 150528
    const int nZ  = KDIM * NDIM;               // 451584
    const int nWt = KDIM * MDIM;               // 73728

    build_t5 <<<(nT5 + 255) / 256, 256, 0, stream>>>(x, W1, T5);
    build_zp <<<(nZ  + 255) / 256, 256, 0, stream>>>(T5, Zp);
    build_w0p<<<(nWt + 255) / 256, 256, 0, stream>>>(W0, W0p);
    gemm_wmma<<<NDIM / 16, 256, 0, stream>>>(W0p, Zp, y);
}